// Net_1_8950711845642
// MI455X (gfx1250) — compile-verified
//
#include <hip/hip_runtime.h>
#include <math.h>

// ---------------------------------------------------------------------------
// GNN: 3x (SAGEConv -> TopKPool -> readout) + MLP head, for MI455X (gfx1250).
// Matmuls use V_WMMA_F32_16X16X4_F32 (full fp32 precision; topk selection is
// sensitive to score rounding, and we are nowhere near the matrix roofline).
// Edge aggregation = global_atomic_add_f32, L2-resident (working set < 192MB).
// ---------------------------------------------------------------------------

typedef __attribute__((ext_vector_type(2))) float v2f;
typedef __attribute__((ext_vector_type(8))) float v8f;

#define F 128            // feature dim (all layers)
#define NGRAPH 64
#define NEDGE (64 * 2048 * 16)

// ---------------- utility fills ----------------
__global__ void fill_f32(float* __restrict__ p, float v, unsigned n) {
  unsigned i = blockIdx.x * blockDim.x + threadIdx.x;
  if (i < n) p[i] = v;
}

__global__ void edge_init(const int* __restrict__ src_in, const int* __restrict__ dst_in,
                          int* __restrict__ src, int* __restrict__ dst,
                          float* __restrict__ em, unsigned nE) {
  unsigned e = blockIdx.x * blockDim.x + threadIdx.x;
  if (e < nE) { src[e] = src_in[e]; dst[e] = dst_in[e]; em[e] = 1.0f; }
}

// ---------------- edge scatter: agg[dst] += x[src]*em ; cnt[dst] += em ------
// 128 threads per edge, one float per thread: coalesced 512B row per edge.
__global__ void scatter_edges(const int* __restrict__ src, const int* __restrict__ dst,
                              const float* __restrict__ em, const float* __restrict__ x,
                              float* __restrict__ agg, float* __restrict__ cnt, unsigned nE) {
  unsigned gid = blockIdx.x * blockDim.x + threadIdx.x;
  unsigned e = gid >> 7;
  unsigned f = gid & 127u;
  if (e >= nE) return;
  float m = em[e];
  if (m != 0.0f) {
    int s = src[e], d = dst[e];
    atomicAdd(&agg[(size_t)d * F + f], x[(size_t)s * F + f] * m);
    if (f == 0) atomicAdd(&cnt[d], m);
  }
}

// ---------------- SAGE gemm (WMMA fp32) ------------------------------------
// h_out = relu( (agg/max(cnt,1)) @ Wl + b + h_in @ Wr )
// grid.x = n_nodes/16 (16-row tile), block = 256 = 8 waves, wave w -> cols [16w,16w+16)
// A 16x4 f32 frag: lane L<16: row=L, {K=k,k+1}; lane>=16: row=L-16, {K=k+2,k+3}
// B 4x16 frag mirrors by column. C/D: vgpr j -> (M=j+8*(L>=16), N=L&15).
__global__ void sage_gemm_relu(const float* __restrict__ agg, const float* __restrict__ cnt,
                               const float* __restrict__ h_in,
                               const float* __restrict__ Wl, const float* __restrict__ bias,
                               const float* __restrict__ Wr,
                               float* __restrict__ h_out, int n_nodes) {
  const int lane = threadIdx.x & 31;
  const int wave = threadIdx.x >> 5;
  const int m0   = blockIdx.x * 16;
  const int n0   = wave * 16;
  const int half = lane >> 4;       // 0 or 1
  const int l16  = lane & 15;
  const int row  = m0 + l16;

  const float inv = 1.0f / fmaxf(cnt[row], 1.0f);
  const float* aRow = agg  + (size_t)row * F;
  const float* xRow = h_in + (size_t)row * F;

  v8f acc = {};
  // mean @ Wl
  #pragma unroll 8
  for (int k = 0; k < F; k += 4) {
    const int ka = k + 2 * half;
    v2f a, b;
    a.x = aRow[ka]     * inv;
    a.y = aRow[ka + 1] * inv;
    b.x = Wl[(size_t)ka       * F + n0 + l16];
    b.y = Wl[(size_t)(ka + 1) * F + n0 + l16];
    acc = __builtin_amdgcn_wmma_f32_16x16x4_f32(false, a, false, b, (short)0, acc, false, false);
  }
  // x @ Wr
  #pragma unroll 8
  for (int k = 0; k < F; k += 4) {
    const int ka = k + 2 * half;
    v2f a, b;
    a.x = xRow[ka];
    a.y = xRow[ka + 1];
    b.x = Wr[(size_t)ka       * F + n0 + l16];
    b.y = Wr[(size_t)(ka + 1) * F + n0 + l16];
    acc = __builtin_amdgcn_wmma_f32_16x16x4_f32(false, a, false, b, (short)0, acc, false, false);
  }

  const float bv = bias[n0 + l16];
  #pragma unroll
  for (int j = 0; j < 8; ++j) {
    float v = acc[j] + bv;
    v = fmaxf(v, 0.0f);
    h_out[(size_t)(m0 + j + 8 * half) * F + (n0 + l16)] = v;
  }
}

// ---------------- topk scoring ---------------------------------------------
__global__ void wnorm_kernel(const float* __restrict__ w, float* __restrict__ inv_norm) {
  __shared__ float s[F];
  int t = threadIdx.x;
  float v = w[t];
  s[t] = v * v;
  __syncthreads();
  for (int off = 64; off > 0; off >>= 1) {
    if (t < off) s[t] += s[t + off];
    __syncthreads();
  }
  if (t == 0) *inv_norm = 1.0f / sqrtf(s[0]);
}

// one wave per node: score = tanh(dot(h, w) * inv_norm)
__global__ void score_kernel(const float* __restrict__ h, const float* __restrict__ w,
                             const float* __restrict__ inv_norm, float* __restrict__ score,
                             int n_nodes) {
  int node = blockIdx.x * (blockDim.x >> 5) + (threadIdx.x >> 5);
  int lane = threadIdx.x & 31;
  if (node >= n_nodes) return;
  const float* r = h + (size_t)node * F;
  float acc = 0.0f;
  #pragma unroll
  for (int k = lane; k < F; k += 32) acc += r[k] * w[k];
  #pragma unroll
  for (int off = 16; off > 0; off >>= 1) acc += __shfl_xor(acc, off, 32);
  if (lane == 0) score[node] = tanhf(acc * (*inv_norm));
}

// rank-by-counting per graph; reproduces lax.top_k order (desc, ties by index)
__global__ void topk_kernel(const float* __restrict__ score, int* __restrict__ kept,
                            int* __restrict__ newidx, int n_per, int k) {
  __shared__ float s[2048];
  int g = blockIdx.x;
  const float* sc = score + (size_t)g * n_per;
  for (int i = threadIdx.x; i < n_per; i += blockDim.x) s[i] = sc[i];
  __syncthreads();
  for (int i = threadIdx.x; i < n_per; i += blockDim.x) {
    float si = s[i];
    int rank = 0;
    for (int j = 0; j < n_per; ++j) {
      float sj = s[j];
      rank += (sj > si) || (sj == si && j < i);
    }
    int node = g * n_per + i;
    if (rank < k) { kept[node] = 1; newidx[node] = g * k + rank; }
    else          { kept[node] = 0; newidx[node] = 0; }
  }
}

// gather kept nodes: h_new[newidx] = h[node] * score[node]   (float4 lanes)
__global__ void pool_apply(const float* __restrict__ h, const float* __restrict__ score,
                           const int* __restrict__ kept, const int* __restrict__ newidx,
                           float* __restrict__ h_new, int n_old_total) {
  unsigned gid = blockIdx.x * blockDim.x + threadIdx.x;
  int node = (int)(gid >> 5);
  int q    = (int)(gid & 31u);
  if (node >= n_old_total) return;
  if (!kept[node]) return;
  float sc = score[node];
  const float4* s4 = (const float4*)(h + (size_t)node * F);
  float4 v = s4[q];
  float4* d4 = (float4*)(h_new + (size_t)newidx[node] * F);
  d4[q] = make_float4(v.x * sc, v.y * sc, v.z * sc, v.w * sc);
}

__global__ void edge_relabel(int* __restrict__ src, int* __restrict__ dst,
                             float* __restrict__ em, const int* __restrict__ kept,
                             const int* __restrict__ newidx, unsigned nE) {
  unsigned e = blockIdx.x * blockDim.x + threadIdx.x;
  if (e >= nE) return;
  int s = src[e], d = dst[e];
  em[e] = em[e] * (float)kept[s] * (float)kept[d];
  src[e] = newidx[s];
  dst[e] = newidx[d];
}

// z[g] += concat(max over nodes, mean over nodes); 128 threads per graph
__global__ void readout_accum(const float* __restrict__ h, float* __restrict__ z, int n_per) {
  int g = blockIdx.x;
  int f = threadIdx.x;
  const float* base = h + (size_t)g * n_per * F + f;
  float mx = -INFINITY, sm = 0.0f;
  for (int i = 0; i < n_per; ++i) {
    float v = base[(size_t)i * F];
    mx = fmaxf(mx, v);
    sm += v;
  }
  z[g * 256 + f]       += mx;
  z[g * 256 + 128 + f] += sm / (float)n_per;
}

// MLP head + log_softmax, one block (256 threads) per graph
__global__ void mlp_head(const float* __restrict__ z,
                         const float* __restrict__ l1w, const float* __restrict__ l1b,
                         const float* __restrict__ l2w, const float* __restrict__ l2b,
                         const float* __restrict__ l3w, const float* __restrict__ l3b,
                         float* __restrict__ out) {
  __shared__ float zr[256];
  __shared__ float h1[128];
  __shared__ float h2[64];
  int g = blockIdx.x, t = threadIdx.x;
  zr[t] = z[g * 256 + t];
  __syncthreads();
  if (t < 128) {
    float a = l1b[t];
    for (int k = 0; k < 256; ++k) a += zr[k] * l1w[k * 128 + t];
    h1[t] = fmaxf(a, 0.0f);
  }
  __syncthreads();
  if (t < 64) {
    float a = l2b[t];
    for (int k = 0; k < 128; ++k) a += h1[k] * l2w[k * 64 + t];
    h2[t] = fmaxf(a, 0.0f);
  }
  __syncthreads();
  if (t == 0) {
    float o0 = l3b[0], o1 = l3b[1];
    for (int k = 0; k < 64; ++k) { o0 += h2[k] * l3w[k * 2 + 0]; o1 += h2[k] * l3w[k * 2 + 1]; }
    float mx  = fmaxf(o0, o1);
    float lse = mx + logf(expf(o0 - mx) + expf(o1 - mx));
    out[g * 2 + 0] = o0 - lse;
    out[g * 2 + 1] = o1 - lse;
  }
}

// ---------------------------------------------------------------------------
static inline char* bump(char*& p, size_t bytes) {
  char* r = p;
  p += (bytes + 255) & ~(size_t)255;
  return r;
}

extern "C" void kernel_launch(void* const* d_in, const int* in_sizes, int n_in,
                              void* d_out, int out_size, void* d_ws, size_t ws_size,
                              hipStream_t stream) {
  (void)in_sizes; (void)n_in; (void)out_size; (void)ws_size;

  const float* x   = (const float*)d_in[0];
  const int*   src = (const int*)d_in[1];
  const int*   dst = (const int*)d_in[2];
  const float* Wl[3] = { (const float*)d_in[3],  (const float*)d_in[7],  (const float*)d_in[11] };
  const float* Bb[3] = { (const float*)d_in[4],  (const float*)d_in[8],  (const float*)d_in[12] };
  const float* Wr[3] = { (const float*)d_in[5],  (const float*)d_in[9],  (const float*)d_in[13] };
  const float* Pw[3] = { (const float*)d_in[6],  (const float*)d_in[10], (const float*)d_in[14] };
  const float* l1w = (const float*)d_in[15]; const float* l1b = (const float*)d_in[16];
  const float* l2w = (const float*)d_in[17]; const float* l2b = (const float*)d_in[18];
  const float* l3w = (const float*)d_in[19]; const float* l3b = (const float*)d_in[20];
  float* out = (float*)d_out;

  const int N0 = 131072;            // nodes entering layer 1
  const unsigned nE = NEDGE;        // 2,097,152 (edges are masked, never removed)

  // -------- workspace carve-up (~190 MB) --------
  char* p = (char*)d_ws;
  float* agg  = (float*)bump(p, (size_t)N0 * F * 4);        // 64 MB, reused all layers
  float* hbuf = (float*)bump(p, (size_t)N0 * F * 4);        // 64 MB, sage outputs
  float* pbuf = (float*)bump(p, (size_t)(N0 / 2) * F * 4);  // 32 MB, pooled features
  int*   srcW = (int*)  bump(p, (size_t)nE * 4);
  int*   dstW = (int*)  bump(p, (size_t)nE * 4);
  float* emW  = (float*)bump(p, (size_t)nE * 4);
  float* cnt  = (float*)bump(p, (size_t)N0 * 4);
  float* scr  = (float*)bump(p, (size_t)N0 * 4);
  int*   kept = (int*)  bump(p, (size_t)N0 * 4);
  int*   nidx = (int*)  bump(p, (size_t)N0 * 4);
  float* invn = (float*)bump(p, 256);
  float* z    = (float*)bump(p, (size_t)NGRAPH * 256 * 4);

  // per-layer shapes
  const int n_nodes[3] = { 131072, 65536, 32768 };  // nodes entering SAGE l
  const int n_per[3]   = { 2048, 1024, 512 };       // nodes/graph entering pool l
  const int kk[3]      = { 1024, 512, 256 };        // kept/graph after pool l

  // init edges + zero z
  {
    unsigned nb = (nE + 255) / 256;
    edge_init<<<nb, 256, 0, stream>>>(src, dst, srcW, dstW, emW, nE);
    fill_f32<<<(NGRAPH * 256 + 255) / 256, 256, 0, stream>>>(z, 0.0f, NGRAPH * 256);
  }

  const float* h_in = x;
  for (int l = 0; l < 3; ++l) {
    const int nn = n_nodes[l];
    // zero agg + cnt
    {
      unsigned na = (unsigned)nn * F;
      fill_f32<<<(na + 255) / 256, 256, 0, stream>>>(agg, 0.0f, na);
      fill_f32<<<((unsigned)nn + 255) / 256, 256, 0, stream>>>(cnt, 0.0f, (unsigned)nn);
    }
    // scatter: agg[dst] += h_in[src]*em ; cnt[dst] += em
    {
      unsigned total = nE * 128u;   // 268M threads max, fits 32-bit grid math
      scatter_edges<<<total / 256, 256, 0, stream>>>(srcW, dstW, emW, h_in, agg, cnt, nE);
    }
    // SAGE gemm + relu  (WMMA fp32)
    sage_gemm_relu<<<nn / 16, 256, 0, stream>>>(agg, cnt, h_in, Wl[l], Bb[l], Wr[l], hbuf, nn);
    // topk scoring
    wnorm_kernel<<<1, 128, 0, stream>>>(Pw[l], invn);
    score_kernel<<<(nn / 8), 256, 0, stream>>>(hbuf, Pw[l], invn, scr, nn);
    topk_kernel<<<NGRAPH, 256, 0, stream>>>(scr, kept, nidx, n_per[l], kk[l]);
    // gather kept, gate by score
    {
      unsigned total = (unsigned)nn * 32u;
      pool_apply<<<(total + 255) / 256, 256, 0, stream>>>(hbuf, scr, kept, nidx, pbuf, nn);
    }
    // relabel edges (not needed after final pool)
    if (l < 2) {
      unsigned nb = (nE + 255) / 256;
      edge_relabel<<<nb, 256, 0, stream>>>(srcW, dstW, emW, kept, nidx, nE);
    }
    // readout of pooled features into z
    readout_accum<<<NGRAPH, 128, 0, stream>>>(pbuf, z, kk[l]);
    h_in = pbuf;   // next layer consumes pooled features
  }

  // MLP head + log_softmax
  mlp_head<<<NGRAPH, 256, 0, stream>>>(z, l1w, l1b, l2w, l2b, l3w, l3b, out);
}